// ComplexSuperposition_53326313947379
// MI455X (gfx1250) — compile-verified
//
#include <hip/hip_runtime.h>

// ComplexSuperposition on MI455X (gfx1250, wave32).
//
// Phase 1: out_r[b,d] = sum_t w[b,t]*x_r[b,t,d]   (64 MB read)
//   - T-reduction split in 2 across blockIdx.z -> 4096 waves, unroll 16
//     => ~16 MB of in-flight reads (matches HBM bandwidth-delay product).
//   - Partials land in d_ws; the combine is folded into phase 2.
// Phase 2: output_r = out_r*out_r^T + out_i*out_i^T
//          output_i = out_i*out_r^T - out_r*out_i^T  (256 MB write)
//   K=2 matmul (zero-padded to K=4) -> one V_WMMA_F32_16X16X4_F32 per
//   16x16 output tile, 2 WMMAs per wave. Operand loads are issued
//   unconditionally (one clause) and lane-masked with selects afterwards,
//   avoiding exec-masked branch chains. Bandwidth floor ~22us total.

typedef __attribute__((ext_vector_type(2))) float v2f;
typedef __attribute__((ext_vector_type(8))) float v8f;

#define NB 128
#define NT 128
#define ND 512

// ---------------- Phase 1: weighted reduction over T (split in 2) --------
// grid = (NB, 2, 2), block = 256. blockIdx.y picks the d-half, blockIdx.z
// picks the t-half. d-loads fully coalesced; weight loads block-uniform.
__global__ __launch_bounds__(256) void phase1_reduce(
    const float* __restrict__ xr, const float* __restrict__ xi,
    const float* __restrict__ w, float* __restrict__ partR,
    float* __restrict__ partI) {
  const int b = blockIdx.x;
  const int d = blockIdx.y * 256 + threadIdx.x;
  const int tz = blockIdx.z;        // which half of T
  const int t0 = tz * (NT / 2);
  const float* __restrict__ wb = w + b * NT + t0;
  const float* __restrict__ pr = xr + ((size_t)b * NT + t0) * ND + d;
  const float* __restrict__ pi = xi + ((size_t)b * NT + t0) * ND + d;
  float ar = 0.f, ai = 0.f;
#pragma unroll 16
  for (int t = 0; t < NT / 2; ++t) {
    const float wt = wb[t];
    ar = fmaf(pr[(size_t)t * ND], wt, ar);
    ai = fmaf(pi[(size_t)t * ND], wt, ai);
  }
  const size_t o = (size_t)tz * NB * ND + (size_t)b * ND + d;
  partR[o] = ar;
  partI[o] = ai;
}

// ---------------- Phase 2: complex outer products via WMMA ----------------
// One wave -> one 16x16 tile of BOTH outputs (2 WMMA ops).
// A (16x4 f32): lanes 0-15 hold M=lane, K={0,1} in {v0,v1}; lanes 16-31
// (K={2,3}) zero.  B (4x16): lanes 0-15 carry rows K=0/K=1; lanes 16-31
// zero.  D: VGPR v = row v (lanes 0-15) / v+8 (lanes 16-31), col = lane&15.
__global__ __launch_bounds__(256) void phase2_outer_wmma(
    const float* __restrict__ partR, const float* __restrict__ partI,
    float* __restrict__ dstR, float* __restrict__ dstI) {
  const int lane = threadIdx.x & 31;
  const int gwave = blockIdx.x * 8 + (threadIdx.x >> 5);
  const int tile = gwave & 1023;  // 32x32 tiles of 16x16 per batch
  const int b = gwave >> 10;
  const int ti = tile >> 5;
  const int tj = tile & 31;
  const int l15 = lane & 15;
  const bool lo = lane < 16;

  const float* __restrict__ r0 = partR + (size_t)b * ND;
  const float* __restrict__ r1 = r0 + (size_t)NB * ND;
  const float* __restrict__ i0 = partI + (size_t)b * ND;
  const float* __restrict__ i1 = i0 + (size_t)NB * ND;

  // Unconditional loads (valid addresses for all lanes): one load clause,
  // one wait. Lane-mask with selects afterwards so EXEC stays full.
  const int ii = ti * 16 + l15;
  const int jj = tj * 16 + l15;
  const float arv = r0[ii] + r1[ii];
  const float aiv = i0[ii] + i1[ii];
  const float brv = r0[jj] + r1[jj];
  const float biv = i0[jj] + i1[jj];
  const float ar = lo ? arv : 0.f;
  const float ai = lo ? aiv : 0.f;
  const float br = lo ? brv : 0.f;
  const float bi = lo ? biv : 0.f;

  v2f Ar = {ar, ai};   // columns: out_r, out_i
  v2f Ai = {ai, -ar};  // columns: out_i, -out_r
  v2f Bv = {br, bi};   // rows:    out_r, out_i
  v8f c = {};

  v8f dr = __builtin_amdgcn_wmma_f32_16x16x4_f32(false, Ar, false, Bv,
                                                 (short)0, c, false, false);
  v8f di = __builtin_amdgcn_wmma_f32_16x16x4_f32(false, Ai, false, Bv,
                                                 (short)0, c, false, false);

  const size_t base = (size_t)b * ND * ND;
  const int col = tj * 16 + l15;
  const int row0 = ti * 16 + (lo ? 0 : 8);
#pragma unroll
  for (int v = 0; v < 8; ++v) {
    const size_t idx = base + (size_t)(row0 + v) * ND + col;
    dstR[idx] = dr[v];
    dstI[idx] = di[v];
  }
}

extern "C" void kernel_launch(void* const* d_in, const int* in_sizes, int n_in,
                              void* d_out, int out_size, void* d_ws,
                              size_t ws_size, hipStream_t stream) {
  const float* xr = (const float*)d_in[0];  // (128,128,512) f32
  const float* xi = (const float*)d_in[1];  // (128,128,512) f32
  const float* w = (const float*)d_in[2];   // (128,128) f32

  float* dstR = (float*)d_out;  // output_r then output_i, flat
  float* dstI = dstR + (size_t)NB * ND * ND;

  // ws: partR[2][NB][ND] then partI[2][NB][ND]  (1 MB total)
  float* partR = (float*)d_ws;
  float* partI = partR + (size_t)2 * NB * ND;

  phase1_reduce<<<dim3(NB, 2, 2), 256, 0, stream>>>(xr, xi, w, partR, partI);

  // 128 batches * 1024 tiles / 8 waves-per-block
  phase2_outer_wmma<<<(NB * 1024) / 8, 256, 0, stream>>>(partR, partI, dstR,
                                                         dstI);
}